// RGBT_Fusion_3_39092792328562
// MI455X (gfx1250) — compile-verified
//
#include <hip/hip_runtime.h>
#include <hip/hip_bf16.h>

typedef __attribute__((ext_vector_type(2))) float v2f;
typedef __attribute__((ext_vector_type(4))) float v4f;
typedef __attribute__((ext_vector_type(8))) float v8f;

#define BATCH 4
#define HH 64
#define WW 64
#define HP 66               // haloed spatial extent
#define CIN 512
#define COUT 256
#define PT 9
#define HW (HH*WW)
#define MTOT (BATCH*HW)     // 16384
#define K1 (PT*CIN)         // 4608
#define K2 (PT*COUT)        // 2304
#define NOFF 32
#define ASTRIDE 68          // LDS row stride (floats): 68 % 64 == 4 -> conflict-free frag reads
#define MTILE 128

__device__ __forceinline__ v8f wmma_f32(v2f a, v2f b, v8f c) {
  // D = A(16x4 f32) * B(4x16 f32) + C(16x16 f32)
  return __builtin_amdgcn_wmma_f32_16x16x4_f32(false, a, false, b, (short)0, c, false, false);
}

// ---------------- zero fill (vectorized) ----------------
__global__ void k_fill0(float* __restrict__ p, int n4) {
  v4f z = {0, 0, 0, 0};
  for (int i = blockIdx.x * blockDim.x + threadIdx.x; i < n4; i += gridDim.x * blockDim.x)
    *(v4f*)(p + (size_t)i * 4) = z;
}

// ---------------- Stage 1: concat + transpose into zero-haloed channels-last ----------------
__global__ void k_pack_x(const float* __restrict__ xv, const float* __restrict__ xi,
                         float* __restrict__ x_hal) {
  __shared__ float tile[32][33];
  int bh = blockIdx.z; int b = bh >> 6; int h = bh & 63;
  int w0 = blockIdx.x * 32, c0 = blockIdx.y * 32;
#pragma unroll
  for (int j = 0; j < 4; ++j) {
    int c = c0 + threadIdx.y + j * 8;
    int w = w0 + threadIdx.x;
    float v = (c < 256) ? xv[((size_t)(b * 256 + c) * HH + h) * WW + w]
                        : xi[((size_t)(b * 256 + (c - 256)) * HH + h) * WW + w];
    tile[threadIdx.y + j * 8][threadIdx.x] = v;
  }
  __syncthreads();
#pragma unroll
  for (int j = 0; j < 4; ++j) {
    int w = w0 + threadIdx.y + j * 8;
    int c = c0 + threadIdx.x;
    x_hal[((size_t)((b * HP + h + 1) * HP + (w + 1))) * CIN + c] = tile[threadIdx.x][threadIdx.y + j * 8];
  }
}

// ---------------- Stage 2: weight repack into B-fragment layout ----------------
// src: [O][Cin][3][3]; k = p*Cin + c (tap-major); layout dst[kq][half][n][g], k = 4*kq + 2*half + g
__global__ void k_pack_w(const float* __restrict__ src, float* __restrict__ dst,
                         int O, int Cin, int padN) {
  int K = 9 * Cin;
  size_t total = (size_t)K * padN;
  for (size_t i = blockIdx.x * (size_t)blockDim.x + threadIdx.x; i < total;
       i += (size_t)gridDim.x * blockDim.x) {
    int k = (int)(i / padN), n = (int)(i % padN);
    int p = k / Cin, c = k % Cin;
    float v = (n < O) ? src[((size_t)n * Cin + c) * 9 + p] : 0.f;
    int kq = k >> 2, hf = (k >> 1) & 1, g = k & 1;
    dst[(((size_t)kq * 2 + hf) * padN + n) * 2 + g] = v;
  }
}

// ---------------- Stage 3: offset conv -> ypos/xpos/mask (branch-free A loads) ----------------
__global__ void k_offset_conv(const float* __restrict__ x_hal,
                              const float* __restrict__ woffp,
                              const float* __restrict__ b_off,
                              float* __restrict__ ypos, float* __restrict__ xpos,
                              float* __restrict__ maskb) {
  int wave = threadIdx.x >> 5, lane = threadIdx.x & 31;
  int lm = lane & 15, half = lane >> 4;
  int m0 = (blockIdx.x * 8 + wave) * 16;
  int b = m0 >> 12, h = (m0 >> 6) & 63, w0 = m0 & 63;
  int w = w0 + lm;
  v8f acc0 = {0, 0, 0, 0, 0, 0, 0, 0};
  v8f acc1 = acc0;
  int kq = 0;
  for (int p = 0; p < PT; ++p) {
    int ky = p / 3, kx = p % 3;
    // halo coords: (h-1+ky)+1 = h+ky ; (w-1+kx)+1 = w+kx  -> always in range, border is zero
    const float* arow = x_hal + ((size_t)((b * HP + h + ky) * HP + (w + kx))) * CIN + half * 2;
    for (int cq = 0; cq < CIN / 4; ++cq, ++kq) {
      v2f a = *(const v2f*)(arow + cq * 4);
      const float* bp = woffp + ((size_t)(kq * 2 + half) * NOFF) * 2;
      v2f b0 = *(const v2f*)(bp + lm * 2);
      v2f b1 = *(const v2f*)(bp + (lm + 16) * 2);
      acc0 = wmma_f32(a, b0, acc0);
      acc1 = wmma_f32(a, b1, acc1);
    }
  }
#pragma unroll
  for (int t = 0; t < 2; ++t) {
    int n = lm + 16 * t;
    if (n >= 27) continue;
    v8f acc = t ? acc1 : acc0;
    float bias = b_off[n];
#pragma unroll
    for (int r = 0; r < 8; ++r) {
      int mm = m0 + half * 8 + r;
      int b2 = mm >> 12, h2 = (mm >> 6) & 63, w2 = mm & 63;
      float v = acc[r] + bias;
      size_t oi;
      if (n < 18) {
        int p = n >> 1;
        oi = (size_t)((b2 * PT + p) * HW) + h2 * WW + w2;
        if ((n & 1) == 0) ypos[oi] = (float)(h2 - 1 + p / 3) + v;
        else              xpos[oi] = (float)(w2 - 1 + p % 3) + v;
      } else {
        int p = n - 18;
        oi = (size_t)((b2 * PT + p) * HW) + h2 * WW + w2;
        maskb[oi] = 1.f / (1.f + __expf(-v));
      }
    }
  }
}

// ---------------- Stage 4: deformable conv implicit GEMM (M-tile 128) ----------------
__global__ void k_dcn(const float* __restrict__ x_hal,
                      const float* __restrict__ ypos, const float* __restrict__ xpos,
                      const float* __restrict__ maskb,
                      const float* __restrict__ wdcnp,
                      float* __restrict__ out1) {
  __shared__ float As[MTILE * ASTRIDE];
  int tid = threadIdx.x;
  int wave = tid >> 5, lane = tid & 31, lm = lane & 15, half = lane >> 4;
  int m0 = blockIdx.x * MTILE;              // 128-aligned -> two full rows
  int b = m0 >> 12;
  int sp0 = m0 & 4095;                      // spatial offset of first pixel (h*64+w)
  int waveN = wave * 32;
  v8f zero8 = {0, 0, 0, 0, 0, 0, 0, 0};
  v8f acc[8][2];
#pragma unroll
  for (int s = 0; s < 8; ++s) { acc[s][0] = zero8; acc[s][1] = zero8; }

  int tm = tid >> 4;          // 0..15
  int tk = (tid & 15) * 4;    // 0..60

  for (int kc = 0; kc < K1 / 64; ++kc) {
    int p = kc >> 3;              // 8 chunks of 64 channels per tap
    int c0 = (kc & 7) * 64;
    __syncthreads();
#pragma unroll
    for (int mb = 0; mb < 8; ++mb) {
      int ml = mb * 16 + tm;
      size_t pi = (size_t)((b * PT + p) * HW) + sp0 + ml;
      float yp = ypos[pi], xp = xpos[pi], mk = maskb[pi];
      float y0f = floorf(yp), x0f = floorf(xp);
      float ly = yp - y0f, lx = xp - x0f;
      int iy = (int)y0f, ix = (int)x0f;
      v4f acc4 = {0, 0, 0, 0};
#pragma unroll
      for (int dy = 0; dy < 2; ++dy) {
        // clamp into halo: clamping to [-1,64] guarantees a zero value for any
        // reference-invalid corner, so no weight masking is needed.
        int yc1 = min(max(iy + dy + 1, 0), HP - 1);
        float wy = dy ? ly : (1.f - ly);
#pragma unroll
        for (int dx = 0; dx < 2; ++dx) {
          int xc1 = min(max(ix + dx + 1, 0), HP - 1);
          float wgt = wy * (dx ? lx : (1.f - lx));
          v4f v = *(const v4f*)(x_hal + ((size_t)((b * HP + yc1) * HP + xc1)) * CIN + c0 + tk);
          acc4 += v * wgt;
        }
      }
      acc4 *= mk;
      *(v4f*)(&As[ml * ASTRIDE + tk]) = acc4;
    }
    __syncthreads();
#pragma unroll 4
    for (int kk = 0; kk < 16; ++kk) {
      const float* bp = wdcnp + ((size_t)((kc * 16 + kk) * 2 + half) * COUT) * 2;
      v2f b0 = *(const v2f*)(bp + (waveN + lm) * 2);
      v2f b1 = *(const v2f*)(bp + (waveN + lm + 16) * 2);
#pragma unroll
      for (int s = 0; s < 8; ++s) {
        v2f a = *(const v2f*)(&As[(s * 16 + lm) * ASTRIDE + kk * 4 + half * 2]);
        acc[s][0] = wmma_f32(a, b0, acc[s][0]);
        acc[s][1] = wmma_f32(a, b1, acc[s][1]);
      }
    }
  }
#pragma unroll
  for (int s = 0; s < 8; ++s) {
    int m = m0 + s * 16 + half * 8;
#pragma unroll
    for (int t = 0; t < 2; ++t) {
      int n = waveN + t * 16 + lm;
#pragma unroll
      for (int r = 0; r < 8; ++r)
        out1[(size_t)(m + r) * COUT + n] = acc[s][t][r];
    }
  }
}

// ---------------- Stage 5/7: deterministic BN statistics ----------------
__global__ void k_bnstats(const float* __restrict__ src, float* __restrict__ psum,
                          float* __restrict__ psumsq) {
  int ch = threadIdx.x;
  int blk = blockIdx.x;
  float s = 0.f, sq = 0.f;
  int pix0 = blk * (MTOT / 128);
  for (int i = 0; i < MTOT / 128; ++i) {
    float v = src[(size_t)(pix0 + i) * COUT + ch];
    s += v; sq += v * v;
  }
  psum[blk * COUT + ch] = s;
  psumsq[blk * COUT + ch] = sq;
}

__global__ void k_bnfinal(const float* __restrict__ psum, const float* __restrict__ psumsq,
                          const float* __restrict__ gamma, const float* __restrict__ beta,
                          float* __restrict__ sarr, float* __restrict__ tarr) {
  int c = threadIdx.x;
  float s = 0.f, sq = 0.f;
  for (int i = 0; i < 128; ++i) { s += psum[i * COUT + c]; sq += psumsq[i * COUT + c]; }
  float mean = s * (1.f / MTOT);
  float var = sq * (1.f / MTOT) - mean * mean;
  float inv = rsqrtf(var + 1e-5f);
  float sc = gamma[c] * inv;
  sarr[c] = sc;
  tarr[c] = beta[c] - mean * sc;
}

// ---------------- Stage 5b: BN1+ReLU into zero-haloed activation buffer ----------------
__global__ void k_bnapply(const float* __restrict__ out1,
                          const float* __restrict__ s1, const float* __restrict__ t1,
                          float* __restrict__ y1_hal) {
  int idx = blockIdx.x * blockDim.x + threadIdx.x;   // over MTOT*COUT
  if (idx >= MTOT * COUT) return;
  int c = idx & 255;
  int m = idx >> 8;
  int b = m >> 12, h = (m >> 6) & 63, w = m & 63;
  float v = fmaxf(out1[(size_t)m * COUT + c] * s1[c] + t1[c], 0.f);
  y1_hal[((size_t)((b * HP + h + 1) * HP + (w + 1))) * COUT + c] = v;
}

// ---------------- Stage 6: conv2 implicit GEMM (A staged via async global->LDS) ----------------
__global__ void k_conv2(const float* __restrict__ y1_hal,
                        const float* __restrict__ w2p,
                        float* __restrict__ out2) {
  __shared__ float As[MTILE * ASTRIDE];
  int tid = threadIdx.x;
  int wave = tid >> 5, lane = tid & 31, lm = lane & 15, half = lane >> 4;
  int m0 = blockIdx.x * MTILE;
  int b = m0 >> 12, h0 = (m0 >> 6) & 63;
  int waveN = wave * 32;
  v8f zero8 = {0, 0, 0, 0, 0, 0, 0, 0};
  v8f acc[8][2];
#pragma unroll
  for (int s = 0; s < 8; ++s) { acc[s][0] = zero8; acc[s][1] = zero8; }

  int tm = tid >> 4;
  int tk = (tid & 15) * 4;

  for (int kc = 0; kc < K2 / 64; ++kc) {
    int p = kc >> 2;              // 4 chunks of 64 channels per tap
    int c0 = (kc & 3) * 64;
    int ky = p / 3, kx = p % 3;
    __syncthreads();
#pragma unroll
    for (int mb = 0; mb < 8; ++mb) {
      int ml = mb * 16 + tm;
      int h = h0 + (ml >> 6);
      int w = ml & 63;
      // halo coords: (h-1+ky)+1 = h+ky ; (w-1+kx)+1 = w+kx
      const float* gp = y1_hal + ((size_t)((b * HP + h + ky) * HP + (w + kx))) * COUT + c0 + tk;
      unsigned ldsa = (unsigned)(size_t)&As[ml * ASTRIDE + tk];   // low 32 bits = LDS offset
      // async DMA copy global -> LDS, 16B per lane, tracked by ASYNCcnt
      asm volatile("global_load_async_to_lds_b128 %0, %1, off"
                   :: "v"(ldsa), "v"(gp)
                   : "memory");
    }
    asm volatile("s_wait_asynccnt 0x0" ::: "memory");  // this wave's LDS writes done
    __syncthreads();                                    // visible to all waves
#pragma unroll 4
    for (int kk = 0; kk < 16; ++kk) {
      const float* bp = w2p + ((size_t)((kc * 16 + kk) * 2 + half) * COUT) * 2;
      v2f b0 = *(const v2f*)(bp + (waveN + lm) * 2);
      v2f b1 = *(const v2f*)(bp + (waveN + lm + 16) * 2);
#pragma unroll
      for (int s = 0; s < 8; ++s) {
        v2f a = *(const v2f*)(&As[(s * 16 + lm) * ASTRIDE + kk * 4 + half * 2]);
        acc[s][0] = wmma_f32(a, b0, acc[s][0]);
        acc[s][1] = wmma_f32(a, b1, acc[s][1]);
      }
    }
  }
#pragma unroll
  for (int s = 0; s < 8; ++s) {
    int m = m0 + s * 16 + half * 8;
#pragma unroll
    for (int t = 0; t < 2; ++t) {
      int n = waveN + t * 16 + lm;
#pragma unroll
      for (int r = 0; r < 8; ++r)
        out2[(size_t)(m + r) * COUT + n] = acc[s][t][r];
    }
  }
}

// ---------------- Stage 8: BN2+ReLU + transpose back to NCHW ----------------
__global__ void k_final(const float* __restrict__ out2,
                        const float* __restrict__ s2, const float* __restrict__ t2,
                        float* __restrict__ out) {
  __shared__ float tile[32][33];
  int b = blockIdx.z;
  int hw0 = blockIdx.x * 32, o0 = blockIdx.y * 32;
#pragma unroll
  for (int j = 0; j < 4; ++j) {
    int hw = hw0 + threadIdx.y + j * 8;
    int o = o0 + threadIdx.x;
    float v = out2[((size_t)(b * HW + hw)) * COUT + o];
    v = fmaxf(v * s2[o] + t2[o], 0.f);
    tile[threadIdx.y + j * 8][threadIdx.x] = v;
  }
  __syncthreads();
#pragma unroll
  for (int j = 0; j < 4; ++j) {
    int o = o0 + threadIdx.y + j * 8;
    int hw = hw0 + threadIdx.x;
    out[((size_t)(b * COUT + o)) * HW + hw] = tile[threadIdx.x][threadIdx.y + j * 8];
  }
}

extern "C" void kernel_launch(void* const* d_in, const int* in_sizes, int n_in,
                              void* d_out, int out_size, void* d_ws, size_t ws_size,
                              hipStream_t stream) {
  const float* input_v = (const float*)d_in[0];
  const float* input_i = (const float*)d_in[1];
  const float* w_off   = (const float*)d_in[2];
  const float* b_off   = (const float*)d_in[3];
  const float* w_dcn   = (const float*)d_in[4];
  const float* gamma1  = (const float*)d_in[5];
  const float* beta1   = (const float*)d_in[6];
  const float* w2      = (const float*)d_in[7];
  const float* gamma2  = (const float*)d_in[8];
  const float* beta2   = (const float*)d_in[9];
  float* out = (float*)d_out;

  float* ws = (float*)d_ws;
  size_t off = 0;
  float* x_hal = ws + off; off += (size_t)BATCH * HP * HP * CIN;   // 8,921,088
  float* ypos  = ws + off; off += (size_t)BATCH * PT * HW;         // 147,456
  float* xpos  = ws + off; off += (size_t)BATCH * PT * HW;
  float* maskb = ws + off; off += (size_t)BATCH * PT * HW;
  float* woffp = ws + off; off += (size_t)K1 * NOFF;               // 147,456
  float* wdcnp = ws + off; off += (size_t)K1 * COUT;               // 1,179,648
  float* w2p   = ws + off; off += (size_t)K2 * COUT;               // 589,824
  float* out1  = ws + off; off += (size_t)MTOT * COUT;             // 4,194,304
  float* y1hal = ws + off; off += (size_t)BATCH * HP * HP * COUT;  // 4,460,544
  float* out2  = ws + off; off += (size_t)MTOT * COUT;             // 4,194,304
  float* psum  = ws + off; off += 128 * COUT;
  float* psumsq= ws + off; off += 128 * COUT;
  float* s1 = ws + off; off += COUT;
  float* t1 = ws + off; off += COUT;
  float* s2 = ws + off; off += COUT;
  float* t2 = ws + off; off += COUT;

  k_fill0<<<2048, 256, 0, stream>>>(x_hal, (int)((size_t)BATCH * HP * HP * CIN / 4));
  k_pack_x<<<dim3(2, 16, BATCH * HH), dim3(32, 8), 0, stream>>>(input_v, input_i, x_hal);
  k_pack_w<<<576, 256, 0, stream>>>(w_off, woffp, 27, CIN, NOFF);
  k_pack_w<<<2048, 256, 0, stream>>>(w_dcn, wdcnp, COUT, CIN, COUT);
  k_pack_w<<<1024, 256, 0, stream>>>(w2, w2p, COUT, COUT, COUT);
  k_offset_conv<<<MTOT / 128, 256, 0, stream>>>(x_hal, woffp, b_off, ypos, xpos, maskb);
  k_dcn<<<MTOT / MTILE, 256, 0, stream>>>(x_hal, ypos, xpos, maskb, wdcnp, out1);
  k_bnstats<<<128, 256, 0, stream>>>(out1, psum, psumsq);
  k_bnfinal<<<1, 256, 0, stream>>>(psum, psumsq, gamma1, beta1, s1, t1);
  k_fill0<<<2048, 256, 0, stream>>>(y1hal, (int)((size_t)BATCH * HP * HP * COUT / 4));
  k_bnapply<<<(MTOT * COUT) / 256, 256, 0, stream>>>(out1, s1, t1, y1hal);
  k_conv2<<<MTOT / MTILE, 256, 0, stream>>>(y1hal, w2p, out2);
  k_bnstats<<<128, 256, 0, stream>>>(out2, psum, psumsq);
  k_bnfinal<<<1, 256, 0, stream>>>(psum, psumsq, gamma2, beta2, s2, t2);
  k_final<<<dim3(HW / 32, COUT / 32, BATCH), dim3(32, 8), 0, stream>>>(out2, s2, t2, out);
}